// KalmanForecasterClosure_62646392980131
// MI455X (gfx1250) — compile-verified
//
#include <hip/hip_runtime.h>
#include <math.h>

// Problem constants (from reference): B=32768, L=512, H=96
#define B_ROWS 32768
#define LHIST  512
#define HFUT   96

constexpr int BLK   = 64;        // threads per block == rows per block (2 wave32s)
constexpr int TC    = 16;        // timesteps per LDS chunk
constexpr int STR   = TC + 4;    // padded LDS row stride in dwords (80B: 16B-aligned, spreads banks)
constexpr int NBUF  = 3;         // async pipeline depth (prefetch 2 chunks ahead)
constexpr int NCH_H = LHIST / TC;  // 32 history chunks
constexpr int NCH_F = HFUT  / TC;  // 6 future chunks

struct KP {
  float a1, b1, b2, c, kappa, vc2, d1, d2, d3;
  float qsqx, qsqu, R, nalog2e;
};

__device__ __forceinline__ float softplusf(float x) {
  return fmaxf(x, 0.0f) + log1pf(__expf(-fabsf(x)));
}

// ---- one predict step: ~37 VALU ops + 1 trans (v_exp_f32) ----
__device__ __forceinline__ void predict(const KP& k, float v, float dv, float dt,
                                        float& x, float& u,
                                        float& pxx, float& pxu, float& puu) {
  float rho = __builtin_amdgcn_exp2f(k.nalog2e * dt);       // exp(-alpha*dt)
  float av = fabsf(v), au = fabsf(u);
  float forcing = fmaxf(fmaf(v, v, -k.vc2), 0.0f);          // relu(v^2 - vc^2)
  float g  = k.a1 + fmaf(k.d1, u, fmaf(k.d2, av, k.d3 * au));
  float cl = fmaf(k.b2, dv, k.b1 * v) - u * g;              // closed-loop term
  float h  = fmaf(k.d1, u, k.d3 * au);
  float f22 = fmaf(-(g + h), dt, rho);                      // rho + dcl_du*dt
  float x_p = fmaf(u, dt, x);
  float inner = fmaf(-k.kappa, x, fmaf(k.c, forcing, cl));
  float u_p = fmaf(dt, inner, rho * u);
  float kdt = k.kappa * dt;
  float fpxx = fmaf(dt, pxu, pxx);
  float fpxu = fmaf(dt, puu, pxu);
  float pxx_n = fmaf(k.qsqx, dt, fmaf(dt, fpxu, fpxx));
  float pxu_n = fmaf(-kdt, fpxx, f22 * fpxu);
  float t1 = fmaf(-kdt, pxx, f22 * pxu);
  float t2 = fmaf(-kdt, pxu, f22 * puu);
  float puu_n = fmaf(k.qsqu, dt, fmaf(-kdt, t1, f22 * t2));
  x = x_p; u = u_p; pxx = pxx_n; pxu = pxu_n; puu = puu_n;
}

// ---- history step: predict + Kalman update (algebraically exact simplification of
//      the Joseph form: omk=R/S, pxx<-omk*pxx, pxu<-omk*pxu, puu<-puu-Ku*pxu) ----
__device__ __forceinline__ void hstep(const KP& k, float vcur, float dtraw, float y,
                                      float& x, float& u,
                                      float& pxx, float& pxu, float& puu,
                                      float& v_prev, float& v_prev2) {
  float dt = fmaxf(dtraw, 1e-6f);
  predict(k, v_prev, v_prev - v_prev2, dt, x, u, pxx, pxu, puu);
  float S  = pxx + k.R;
  float rS = __builtin_amdgcn_rcpf(S);                      // v_rcp_f32
  float Kx = pxx * rS, Ku = pxu * rS;
  float innov = y - x;
  x = fmaf(Kx, innov, x);
  u = fmaf(Ku, innov, u);
  float omk = k.R * rS;                                     // == 1-Kx exactly
  float puu_n = fmaf(-Ku, pxu, puu);
  pxx = omk * pxx;
  pxu = omk * pxu;
  puu = puu_n;
  v_prev2 = v_prev; v_prev = vcur;
}

__device__ __forceinline__ void fstep(const KP& k, float v, float dtraw,
                                      float& x, float& u,
                                      float& pxx, float& pxu, float& puu,
                                      float& v_prev,
                                      float& ox, float& op, float& ou) {
  float dt = fmaxf(dtraw, 1e-6f);
  predict(k, v, v - v_prev, dt, x, u, pxx, pxu, puu);
  v_prev = v;
  ox = x; op = pxx; ou = u;
}

// ---- gfx1250 async global->LDS staging: 4x b128 per thread = one 64-row x 16-step tile.
//      Lane quads read 64B contiguous per row -> coalesced; writes land in padded LDS rows.
__device__ __forceinline__ void async_tile(const float* g, int rowlen, int row0,
                                           int t0, const float* lds_arr, int tid) {
#pragma unroll
  for (int p4 = 0; p4 < 4; ++p4) {
    int r  = (tid >> 2) + p4 * (BLK / 4);
    int t4 = (tid & 3) << 2;
    unsigned long long ga =
        (unsigned long long)(const void*)(g + (size_t)(row0 + r) * rowlen + (t0 + t4));
    unsigned la = (unsigned)(size_t)(const void*)(lds_arr + r * STR + t4);
    asm volatile("global_load_async_to_lds_b128 %0, %1, off"
                 :: "v"(la), "v"(ga) : "memory");
  }
}

__device__ __forceinline__ void wait_async_le(int n) {
  if (n >= 24)      asm volatile("s_wait_asynccnt 24" ::: "memory");
  else if (n >= 16) asm volatile("s_wait_asynccnt 16" ::: "memory");
  else if (n >= 12) asm volatile("s_wait_asynccnt 12" ::: "memory");
  else if (n >= 8)  asm volatile("s_wait_asynccnt 8"  ::: "memory");
  else              asm volatile("s_wait_asynccnt 0"  ::: "memory");
}

__global__ void __launch_bounds__(BLK)
kf_kernel(const float* __restrict__ params,
          const float* __restrict__ v_hist,
          const float* __restrict__ dt_hist,
          const float* __restrict__ x_obs,
          const float* __restrict__ v_fut,
          const float* __restrict__ dt_fut,
          float* __restrict__ out) {
  __shared__ __align__(16) float sh[NBUF][3][BLK * STR];   // 45 KB

  const int tid  = threadIdx.x;
  const int row0 = blockIdx.x * BLK;
  const int b    = row0 + tid;

  // ---- derived parameters (uniform across all lanes; computed once) ----
  float p[16];
#pragma unroll
  for (int i = 0; i < 16; ++i) p[i] = params[i];
  KP k;
  float alpha = 1.0f / (1.0f + __expf(-p[0]));
  k.c     = softplusf(p[1]);
  k.kappa = softplusf(p[2]);
  float vc = softplusf(p[3]);
  k.vc2 = vc * vc;
  float qx = __expf(p[4]), qu = __expf(p[5]);
  k.R = __expf(p[6]);
  float qs = __expf(p[7]);
  float p0xx = __expf(p[8]), p0uu = __expf(p[9]);
  k.a1 = softplusf(p[10]);
  k.d1 = softplusf(p[11]);
  k.d2 = softplusf(p[12]);
  k.d3 = softplusf(p[13]);
  k.b1 = p[14]; k.b2 = p[15];
  k.qsqx = qs * qx; k.qsqu = qs * qu;
  k.nalog2e = -alpha * 1.44269504088896340736f;            // -alpha * log2(e)

  // =================== history: 32 chunks, triple-buffered async pipeline ===================
#pragma unroll
  for (int c0 = 0; c0 < NBUF; ++c0) {
    async_tile(v_hist,  LHIST, row0, c0 * TC, &sh[c0][0][0], tid);
    async_tile(dt_hist, LHIST, row0, c0 * TC, &sh[c0][1][0], tid);
    async_tile(x_obs,   LHIST, row0, c0 * TC, &sh[c0][2][0], tid);
  }

  float x = 0.f, u = 0.f, pxx = 0.f, pxu = 0.f, puu = 0.f;
  float v_prev = 0.f, v_prev2 = 0.f;
  int buf = 0;
  for (int c = 0; c < NCH_H; ++c) {
    int rem = NCH_H - 1 - c;
    wait_async_le(12 * (rem < NBUF - 1 ? rem : NBUF - 1));
    __syncthreads();
    const float4* sv4 = (const float4*)(&sh[buf][0][tid * STR]);
    const float4* sd4 = (const float4*)(&sh[buf][1][tid * STR]);
    const float4* sy4 = (const float4*)(&sh[buf][2][tid * STR]);
    if (c == 0) {
      float4 v4 = sv4[0], d4 = sd4[0], y4 = sy4[0];
      x = y4.x; u = 0.f; pxx = p0xx; pxu = 0.f; puu = p0uu;  // carry0
      v_prev = v4.x; v_prev2 = v4.x;                         // dv[0] = 0
      hstep(k, v4.y, d4.y, y4.y, x, u, pxx, pxu, puu, v_prev, v_prev2);
      hstep(k, v4.z, d4.z, y4.z, x, u, pxx, pxu, puu, v_prev, v_prev2);
      hstep(k, v4.w, d4.w, y4.w, x, u, pxx, pxu, puu, v_prev, v_prev2);
#pragma unroll
      for (int k4 = 1; k4 < 4; ++k4) {
        float4 a = sv4[k4], dd = sd4[k4], yy = sy4[k4];
        hstep(k, a.x, dd.x, yy.x, x, u, pxx, pxu, puu, v_prev, v_prev2);
        hstep(k, a.y, dd.y, yy.y, x, u, pxx, pxu, puu, v_prev, v_prev2);
        hstep(k, a.z, dd.z, yy.z, x, u, pxx, pxu, puu, v_prev, v_prev2);
        hstep(k, a.w, dd.w, yy.w, x, u, pxx, pxu, puu, v_prev, v_prev2);
      }
    } else {
#pragma unroll
      for (int k4 = 0; k4 < 4; ++k4) {
        float4 a = sv4[k4], dd = sd4[k4], yy = sy4[k4];
        hstep(k, a.x, dd.x, yy.x, x, u, pxx, pxu, puu, v_prev, v_prev2);
        hstep(k, a.y, dd.y, yy.y, x, u, pxx, pxu, puu, v_prev, v_prev2);
        hstep(k, a.z, dd.z, yy.z, x, u, pxx, pxu, puu, v_prev, v_prev2);
        hstep(k, a.w, dd.w, yy.w, x, u, pxx, pxu, puu, v_prev, v_prev2);
      }
    }
    __syncthreads();                                       // everyone done reading buf
    int cn = c + NBUF;
    if (cn < NCH_H) {
      async_tile(v_hist,  LHIST, row0, cn * TC, &sh[buf][0][0], tid);
      async_tile(dt_hist, LHIST, row0, cn * TC, &sh[buf][1][0], tid);
      async_tile(x_obs,   LHIST, row0, cn * TC, &sh[buf][2][0], tid);
    }
    buf = (buf + 1 == NBUF) ? 0 : buf + 1;
  }

  // =================== future: 6 chunks, same pipeline; outputs streamed as float4 ===================
#pragma unroll
  for (int c0 = 0; c0 < NBUF; ++c0) {
    async_tile(v_fut,  HFUT, row0, c0 * TC, &sh[c0][0][0], tid);
    async_tile(dt_fut, HFUT, row0, c0 * TC, &sh[c0][1][0], tid);
  }
  float* out_x = out;
  float* out_p = out + (size_t)B_ROWS * HFUT;
  float* out_u = out + 2 * (size_t)B_ROWS * HFUT;
  buf = 0;
  for (int c = 0; c < NCH_F; ++c) {
    int rem = NCH_F - 1 - c;
    wait_async_le(8 * (rem < NBUF - 1 ? rem : NBUF - 1));
    __syncthreads();
    const float4* sv4 = (const float4*)(&sh[buf][0][tid * STR]);
    const float4* sd4 = (const float4*)(&sh[buf][1][tid * STR]);
#pragma unroll
    for (int k4 = 0; k4 < 4; ++k4) {
      float4 a = sv4[k4], dd = sd4[k4];
      float rx[4], rp[4], ru[4];
      fstep(k, a.x, dd.x, x, u, pxx, pxu, puu, v_prev, rx[0], rp[0], ru[0]);
      fstep(k, a.y, dd.y, x, u, pxx, pxu, puu, v_prev, rx[1], rp[1], ru[1]);
      fstep(k, a.z, dd.z, x, u, pxx, pxu, puu, v_prev, rx[2], rp[2], ru[2]);
      fstep(k, a.w, dd.w, x, u, pxx, pxu, puu, v_prev, rx[3], rp[3], ru[3]);
      size_t o = (size_t)b * HFUT + (c * TC + k4 * 4);
      *(float4*)(out_x + o) = make_float4(rx[0], rx[1], rx[2], rx[3]);
      *(float4*)(out_p + o) = make_float4(rp[0], rp[1], rp[2], rp[3]);
      *(float4*)(out_u + o) = make_float4(ru[0], ru[1], ru[2], ru[3]);
    }
    __syncthreads();
    int cn = c + NBUF;
    if (cn < NCH_F) {
      async_tile(v_fut,  HFUT, row0, cn * TC, &sh[buf][0][0], tid);
      async_tile(dt_fut, HFUT, row0, cn * TC, &sh[buf][1][0], tid);
    }
    buf = (buf + 1 == NBUF) ? 0 : buf + 1;
  }
}

extern "C" void kernel_launch(void* const* d_in, const int* in_sizes, int n_in,
                              void* d_out, int out_size, void* d_ws, size_t ws_size,
                              hipStream_t stream) {
  const float* params  = (const float*)d_in[0];
  const float* v_hist  = (const float*)d_in[1];
  const float* dt_hist = (const float*)d_in[2];
  const float* x_obs   = (const float*)d_in[3];
  const float* v_fut   = (const float*)d_in[4];
  const float* dt_fut  = (const float*)d_in[5];
  dim3 grid(B_ROWS / BLK), block(BLK);
  hipLaunchKernelGGL(kf_kernel, grid, block, 0, stream,
                     params, v_hist, dt_hist, x_obs, v_fut, dt_fut, (float*)d_out);
}